// FocalLoss3d_ver1_14809047237031
// MI455X (gfx1250) — compile-verified
//
#include <hip/hip_runtime.h>

typedef __attribute__((ext_vector_type(2))) float v2f;
typedef __attribute__((ext_vector_type(8))) float v8f;

#define NB 2
#define ND 128
#define NH 256
#define NW 256
#define NROWS (NB * ND * NH)          // 65536 rows of W=256
#define NTOT  (NB * ND * NH * NW)     // 16777216 voxels
#define NWORDS (NROWS * 8)            // 8 u32 mask words per row
#define PW 10.0f

// ---------------------------------------------------------------------------
// Wave-level sum of 32 lane partials via V_WMMA_F32_16X16X4_F32.
// A (16x4): VGPR0 lanes0-15 = A[m][0]=p_m ; lanes16-31 = A[m][2]=p_{m+16};
// VGPR1 (K=1/K=3) = 0. B = all ones 4x16. D[m][n] = p_m + p_{m+16}.
// C/D layout: lane n<16 holds D[r][n] in VGPR r; lane n>=16 holds D[8+r][n-16].
// sum(col 0) = (sum of 8 regs @lane0) + (sum of 8 regs @lane16) = total.
// Requires EXEC all-ones: call only with full wave converged.
// ---------------------------------------------------------------------------
__device__ inline float wave_reduce_wmma(float p) {
    v2f a; a.x = p;    a.y = 0.0f;
    v2f b; b.x = 1.0f; b.y = 1.0f;
    v8f c = {};
    c = __builtin_amdgcn_wmma_f32_16x16x4_f32(false, a, false, b,
                                              (short)0, c, false, false);
    float s = c[0] + c[1] + c[2] + c[3] + c[4] + c[5] + c[6] + c[7];
    float t0  = __shfl(s, 0, 32);
    float t16 = __shfl(s, 16, 32);
    return t0 + t16;   // full wave sum, broadcast to all lanes
}

__device__ inline float block_reduce(float p, float* lds8) {
    float w = wave_reduce_wmma(p);
    int lane = threadIdx.x & 31;
    int wv   = threadIdx.x >> 5;
    if (lane == 0) lds8[wv] = w;
    __syncthreads();
    float r = 0.0f;
    if (threadIdx.x == 0) {
        #pragma unroll
        for (int i = 0; i < 8; ++i) r += lds8[i];
    }
    return r;  // valid on thread 0
}

// ---------------------------------------------------------------------------
// K1: per row (b,d,h): bitmask of (0 < t < 1) over W=256, dilated +-3 along W.
// One wave per row; each lane covers 8 consecutive w via two b128 loads;
// 256-bit row mask assembled with shuffles, dilated with shifts + carries.
// ---------------------------------------------------------------------------
__global__ void mask_w_kernel(const float* __restrict__ tgt,
                              unsigned int* __restrict__ dst) {
    const int lane   = threadIdx.x & 31;
    const int wave   = (blockIdx.x * blockDim.x + threadIdx.x) >> 5;
    const int nwaves = (gridDim.x * blockDim.x) >> 5;
    for (int row = wave; row < NROWS; row += nwaves) {
        const float4* rp = (const float4*)(tgt + (size_t)row * NW);
        float4 a = rp[lane * 2 + 0];
        float4 b = rp[lane * 2 + 1];
        unsigned int byte = 0;
        byte |= (a.x > 0.0f && a.x < 1.0f) ?   1u : 0u;
        byte |= (a.y > 0.0f && a.y < 1.0f) ?   2u : 0u;
        byte |= (a.z > 0.0f && a.z < 1.0f) ?   4u : 0u;
        byte |= (a.w > 0.0f && a.w < 1.0f) ?   8u : 0u;
        byte |= (b.x > 0.0f && b.x < 1.0f) ?  16u : 0u;
        byte |= (b.y > 0.0f && b.y < 1.0f) ?  32u : 0u;
        byte |= (b.z > 0.0f && b.z < 1.0f) ?  64u : 0u;
        byte |= (b.w > 0.0f && b.w < 1.0f) ? 128u : 0u;
        // lane j (j = lane&7) assembles mask word j from bytes at lanes 4j..4j+3
        int wsel = (lane & 7) * 4;
        unsigned int w0 = __shfl(byte, wsel + 0, 32);
        unsigned int w1 = __shfl(byte, wsel + 1, 32);
        unsigned int w2 = __shfl(byte, wsel + 2, 32);
        unsigned int w3 = __shfl(byte, wsel + 3, 32);
        unsigned int word = w0 | (w1 << 8) | (w2 << 16) | (w3 << 24);
        // neighbor words for cross-word dilation carries
        unsigned int prev = __shfl(word, (lane == 0)  ? 0  : lane - 1, 32);
        unsigned int next = __shfl(word, (lane == 31) ? 31 : lane + 1, 32);
        if (lane == 0) prev = 0u;
        if (lane == 7) next = 0u;
        unsigned int d = word
            | (word << 1) | (word << 2) | (word << 3)
            | (word >> 1) | (word >> 2) | (word >> 3)
            | (prev >> 31) | (prev >> 30) | (prev >> 29)
            | (next << 31) | (next << 30) | (next << 29);
        if (lane < 8) dst[(size_t)row * 8 + lane] = d;
    }
}

// K2: dilate +-3 along H (clamped to [0,256)). One thread per (row, word).
__global__ void dilate_h_kernel(const unsigned int* __restrict__ src,
                                unsigned int* __restrict__ dst) {
    int t = blockIdx.x * blockDim.x + threadIdx.x;  // 0 .. NWORDS-1
    int j  = t & 7;
    int r  = t >> 3;          // (b*D + d)*H + h
    int h  = r & (NH - 1);
    int bd = r >> 8;
    int h0 = h - 3; if (h0 < 0) h0 = 0;
    int h1 = h + 3; if (h1 > NH - 1) h1 = NH - 1;
    unsigned int acc = 0u;
    for (int hh = h0; hh <= h1; ++hh)
        acc |= src[(size_t)(((bd << 8) + hh) * 8 + j)];
    dst[t] = acc;
}

// K3: dilate +-3 along D (clamped to [0,128), within batch).
__global__ void dilate_d_kernel(const unsigned int* __restrict__ src,
                                unsigned int* __restrict__ dst) {
    int t = blockIdx.x * blockDim.x + threadIdx.x;
    int j   = t & 7;
    int r   = t >> 3;
    int h   = r & (NH - 1);
    int bdD = r >> 8;          // b*D + d
    int d   = bdD & (ND - 1);
    int b   = bdD >> 7;
    int d0 = d - 3; if (d0 < 0) d0 = 0;
    int d1 = d + 3; if (d1 > ND - 1) d1 = ND - 1;
    unsigned int acc = 0u;
    for (int dd = d0; dd <= d1; ++dd)
        acc |= src[(size_t)(((((b << 7) + dd) << 8) + h) * 8 + j)];
    dst[t] = acc;
}

// ---------------------------------------------------------------------------
// K4: streaming pass. float4 (b128) loads of input & target; weight from mask
// bit; per-thread partial -> WMMA wave reduce -> per-block partial.
// Uniform trip count => all waves fully converged at the WMMA.
// ---------------------------------------------------------------------------
__global__ void main_loss_kernel(const float* __restrict__ inp,
                                 const float* __restrict__ tgt,
                                 const unsigned int* __restrict__ mask,
                                 float* __restrict__ partials) {
    const int NG = NTOT / 4;                       // 4194304 float4 groups
    int stride = gridDim.x * blockDim.x;
    float acc = 0.0f;
    for (int g = blockIdx.x * blockDim.x + threadIdx.x; g < NG; g += stride) {
        int base = g << 2;
        // mask word index = row*8 + ((base>>5)&7) == base>>5 (words contiguous per row)
        unsigned int word = mask[(unsigned)base >> 5];
        int sh = base & 31;                        // multiple of 4, <= 28
        unsigned int bits = word >> sh;
        const float4 x = ((const float4*)inp)[g];
        const float4 t = ((const float4*)tgt)[g];
        acc += fabsf(t.x - x.x) * (1.0f + PW * (float)( bits        & 1u));
        acc += fabsf(t.y - x.y) * (1.0f + PW * (float)((bits >> 1)  & 1u));
        acc += fabsf(t.z - x.z) * (1.0f + PW * (float)((bits >> 2)  & 1u));
        acc += fabsf(t.w - x.w) * (1.0f + PW * (float)((bits >> 3)  & 1u));
    }
    __shared__ float lds8[8];
    float r = block_reduce(acc, lds8);
    if (threadIdx.x == 0) partials[blockIdx.x] = r;
}

// K5: single-block deterministic reduce of block partials, scale by 1/N.
__global__ void final_reduce_kernel(const float* __restrict__ partials, int n,
                                    float* __restrict__ out, float invN) {
    float acc = 0.0f;
    for (int i = threadIdx.x; i < n; i += blockDim.x) acc += partials[i];
    __shared__ float lds8[8];
    float r = block_reduce(acc, lds8);
    if (threadIdx.x == 0) out[0] = r * invN;
}

extern "C" void kernel_launch(void* const* d_in, const int* in_sizes, int n_in,
                              void* d_out, int out_size, void* d_ws, size_t ws_size,
                              hipStream_t stream) {
    const float* inp = (const float*)d_in[0];
    const float* tgt = (const float*)d_in[1];
    unsigned int* ws0 = (unsigned int*)d_ws;        // 2 MB: W-dilated mask / final mask
    unsigned int* ws1 = ws0 + NWORDS;               // 2 MB: H-dilated mask
    float* partials   = (float*)(ws0 + 2 * NWORDS); // 8 KB: block partials

    mask_w_kernel   <<<1024, 256, 0, stream>>>(tgt, ws0);
    dilate_h_kernel <<<NWORDS / 256, 256, 0, stream>>>(ws0, ws1);
    dilate_d_kernel <<<NWORDS / 256, 256, 0, stream>>>(ws1, ws0);
    main_loss_kernel<<<2048, 256, 0, stream>>>(inp, tgt, ws0, partials);
    final_reduce_kernel<<<1, 256, 0, stream>>>(partials, 2048, (float*)d_out,
                                               1.0f / (float)NTOT);
}